// Plane_Height_loss_52080773431709
// MI455X (gfx1250) — compile-verified
//
#include <hip/hip_runtime.h>
#include <hip/hip_bf16.h>

typedef float vf4 __attribute__((ext_vector_type(4)));
typedef float v2f __attribute__((ext_vector_type(2)));
typedef float v8f __attribute__((ext_vector_type(8)));

#define IMG_W 1280
#define IMG_H 384
#define NPIX  (IMG_W * IMG_H)
#define NBATCH 16
#define BLOCKS_X 128
#define THREADS 256

// ---------------------------------------------------------------------------
// Stage 0: zero the 16x16 float partial-sum table in workspace.
// ---------------------------------------------------------------------------
__global__ void ph_zero_ws(float* __restrict__ ws) {
    ws[threadIdx.x] = 0.0f;
}

// ---------------------------------------------------------------------------
// Stage 1: streaming moment reduction (HBM-bound, ~2.7us floor).
// Accumulators (per batch):
//   a0=S(w)  a1=S(u)  a2=S(u*j)  a3=S(u*i)
//   a4=S(u2) a5=S(u2*j) a6=S(u2*i) a7=S(u2*j^2) a8=S(u2*j*i) a9=S(u2*i^2)
// with u = w*d, u2 = w*d^2, w = plane_est^3.  K^-1 is folded in later.
// Block-level combine of the 8 wave partials is done with
// V_WMMA_F32_16X16X4_F32 using an all-ones A matrix (exact fp32 column sums).
// ---------------------------------------------------------------------------
__global__ __launch_bounds__(THREADS)
void ph_moments(const float* __restrict__ plane_est,
                const float* __restrict__ depth,
                float* __restrict__ ws) {
    const int batch = blockIdx.y;
    const float* peb = plane_est + (size_t)batch * NPIX;
    const float* db  = depth     + (size_t)batch * NPIX;

    float a0 = 0.f, a1 = 0.f, a2 = 0.f, a3 = 0.f, a4 = 0.f;
    float a5 = 0.f, a6 = 0.f, a7 = 0.f, a8 = 0.f, a9 = 0.f;

    const int tid = threadIdx.x;
    const int stride = gridDim.x * THREADS * 4;

#define PH_PIX(WV, DV, FJ, FI)                                   \
    {                                                            \
        float w  = (WV) * (WV) * (WV);                           \
        float u  = w * (DV);                                     \
        float u2 = u * (DV);                                     \
        a0 += w;                                                 \
        a1 += u;                                                 \
        a2 = fmaf(u,  (FJ), a2);                                 \
        a3 = fmaf(u,  (FI), a3);                                 \
        a4 += u2;                                                \
        a5 = fmaf(u2, (FJ), a5);                                 \
        a6 = fmaf(u2, (FI), a6);                                 \
        float tj = u2 * (FJ);                                    \
        float ti = u2 * (FI);                                    \
        a7 = fmaf(tj, (FJ), a7);                                 \
        a8 = fmaf(tj, (FI), a8);                                 \
        a9 = fmaf(ti, (FI), a9);                                 \
    }

    for (int p = (blockIdx.x * THREADS + tid) * 4; p < NPIX; p += stride) {
        // IMG_W % 4 == 0 and p % 4 == 0 -> all 4 pixels share a row.
        const int   i  = p / IMG_W;
        const int   j0 = p - i * IMG_W;
        const float fi = (float)i;
        const float fj = (float)j0;

        // Read-once streams: non-temporal 128-bit loads.
        vf4 d4 = __builtin_nontemporal_load((const vf4*)(db  + p));
        vf4 w4 = __builtin_nontemporal_load((const vf4*)(peb + p));

        PH_PIX(w4.x, d4.x, fj + 0.0f, fi);
        PH_PIX(w4.y, d4.y, fj + 1.0f, fi);
        PH_PIX(w4.z, d4.z, fj + 2.0f, fi);
        PH_PIX(w4.w, d4.w, fj + 3.0f, fi);
    }
#undef PH_PIX

    // --- wave32 butterfly: every lane ends with the wave totals -----------
#define PH_RED(M)                                                \
    a0 += __shfl_xor(a0, (M), 32); a1 += __shfl_xor(a1, (M), 32);\
    a2 += __shfl_xor(a2, (M), 32); a3 += __shfl_xor(a3, (M), 32);\
    a4 += __shfl_xor(a4, (M), 32); a5 += __shfl_xor(a5, (M), 32);\
    a6 += __shfl_xor(a6, (M), 32); a7 += __shfl_xor(a7, (M), 32);\
    a8 += __shfl_xor(a8, (M), 32); a9 += __shfl_xor(a9, (M), 32);
    PH_RED(16) PH_RED(8) PH_RED(4) PH_RED(2) PH_RED(1)
#undef PH_RED

    __shared__ float P[8][16];   // [wave][accumulator], cols 10..15 padded 0
    const int wave = tid >> 5;
    const int lane = tid & 31;
    if (lane == 0) {
        P[wave][0] = a0; P[wave][1] = a1; P[wave][2] = a2; P[wave][3] = a3;
        P[wave][4] = a4; P[wave][5] = a5; P[wave][6] = a6; P[wave][7] = a7;
        P[wave][8] = a8; P[wave][9] = a9;
        P[wave][10] = 0.f; P[wave][11] = 0.f; P[wave][12] = 0.f;
        P[wave][13] = 0.f; P[wave][14] = 0.f; P[wave][15] = 0.f;
    }
    __syncthreads();

    if (wave == 0) {                 // wave-uniform branch: EXEC all ones
        const int h = lane >> 4;     // lane half selects a K row
        const int n = lane & 15;     // column = accumulator index
        v2f aones = {1.0f, 1.0f};    // A(16x4) all ones -> D(m,n)=sum_k B(k,n)
        v2f b0 = { P[0 + h][n], P[2 + h][n] };   // waves 0..3
        v2f b1 = { P[4 + h][n], P[6 + h][n] };   // waves 4..7
        v8f c = {0.f, 0.f, 0.f, 0.f, 0.f, 0.f, 0.f, 0.f};
        c = __builtin_amdgcn_wmma_f32_16x16x4_f32(false, aones, false, b0,
                                                  (short)0, c, false, false);
        c = __builtin_amdgcn_wmma_f32_16x16x4_f32(false, aones, false, b1,
                                                  (short)0, c, false, false);
        // C layout: VGPR0 / lanes 0-15 = (M=0, N=lane); all D rows identical.
        if (lane < 10) {
            atomicAdd(&ws[batch * 16 + lane], c[0]);
        }
    }
}

// ---------------------------------------------------------------------------
// Stage 2: per-batch 3x3 algebra + scale + loss.  One wave.
//   L = Kinv*l,  M = Kinv*Q*Kinv^T,  normal = L*M^-1,  n = normal/|normal|
//   height = (L.n)/S,  s = (1.7/30)/height,  loss = mean_b |s-1|/s
// ---------------------------------------------------------------------------
__global__ void ph_finalize(const float* __restrict__ ws,
                            const float* __restrict__ kinv,
                            float* __restrict__ out) {
    const int b = threadIdx.x;
    float lb = 0.0f;
    if (b < NBATCH) {
        float m[10];
#pragma unroll
        for (int q = 0; q < 10; ++q) m[q] = ws[b * 16 + q];

        float K[3][3];
#pragma unroll
        for (int r = 0; r < 3; ++r)
#pragma unroll
            for (int c = 0; c < 3; ++c) K[r][c] = kinv[b * 9 + r * 3 + c];

        // l in (j,i,1) basis; Q = second-moment matrix in (j,i,1) basis
        float lv[3] = { m[2], m[3], m[1] };
        float Q[3][3] = { { m[7], m[8], m[5] },
                          { m[8], m[9], m[6] },
                          { m[5], m[6], m[4] } };

        float L[3];
#pragma unroll
        for (int r = 0; r < 3; ++r)
            L[r] = K[r][0]*lv[0] + K[r][1]*lv[1] + K[r][2]*lv[2];

        float T[3][3], M[3][3];
#pragma unroll
        for (int r = 0; r < 3; ++r)
#pragma unroll
            for (int c = 0; c < 3; ++c)
                T[r][c] = K[r][0]*Q[0][c] + K[r][1]*Q[1][c] + K[r][2]*Q[2][c];
#pragma unroll
        for (int r = 0; r < 3; ++r)
#pragma unroll
            for (int c = 0; c < 3; ++c)
                M[r][c] = T[r][0]*K[c][0] + T[r][1]*K[c][1] + T[r][2]*K[c][2];

        // inverse of (generically PD) symmetric 3x3 via adjugate
        float c00 = M[1][1]*M[2][2] - M[1][2]*M[2][1];
        float c01 = M[1][2]*M[2][0] - M[1][0]*M[2][2];
        float c02 = M[1][0]*M[2][1] - M[1][1]*M[2][0];
        float det = M[0][0]*c00 + M[0][1]*c01 + M[0][2]*c02;
        float id  = 1.0f / det;
        float inv[3][3];
        inv[0][0] = c00 * id;
        inv[1][0] = c01 * id;
        inv[2][0] = c02 * id;
        inv[0][1] = (M[0][2]*M[2][1] - M[0][1]*M[2][2]) * id;
        inv[1][1] = (M[0][0]*M[2][2] - M[0][2]*M[2][0]) * id;
        inv[2][1] = (M[0][1]*M[2][0] - M[0][0]*M[2][1]) * id;
        inv[0][2] = (M[0][1]*M[1][2] - M[0][2]*M[1][1]) * id;
        inv[1][2] = (M[0][2]*M[1][0] - M[0][0]*M[1][2]) * id;
        inv[2][2] = (M[0][0]*M[1][1] - M[0][1]*M[1][0]) * id;

        float nr[3];
#pragma unroll
        for (int d = 0; d < 3; ++d)
            nr[d] = L[0]*inv[0][d] + L[1]*inv[1][d] + L[2]*inv[2][d];
        float nl = sqrtf(nr[0]*nr[0] + nr[1]*nr[1] + nr[2]*nr[2]);
        float height = (L[0]*nr[0] + L[1]*nr[1] + L[2]*nr[2]) / (nl * m[0]);
        float s = (1.7f / 30.0f) / height;
        out[b] = s;
        lb = fabsf(s - 1.0f) / s;   // per-pixel loss term is batch-constant
    }
    lb += __shfl_xor(lb, 16, 32);
    lb += __shfl_xor(lb, 8, 32);
    lb += __shfl_xor(lb, 4, 32);
    lb += __shfl_xor(lb, 2, 32);
    lb += __shfl_xor(lb, 1, 32);
    if (b == 0) out[NBATCH] = lb * (1.0f / (float)NBATCH);
}

// ---------------------------------------------------------------------------
extern "C" void kernel_launch(void* const* d_in, const int* in_sizes, int n_in,
                              void* d_out, int out_size, void* d_ws, size_t ws_size,
                              hipStream_t stream) {
    const float* plane_est = (const float*)d_in[0];
    const float* depth     = (const float*)d_in[1];
    const float* kinv      = (const float*)d_in[2];
    // d_in[3] (disp) is unused by the reference computation.
    float* out = (float*)d_out;
    float* ws  = (float*)d_ws;   // 16 batches x 16 floats of partial sums

    ph_zero_ws<<<1, NBATCH * 16, 0, stream>>>(ws);
    ph_moments<<<dim3(BLOCKS_X, NBATCH), THREADS, 0, stream>>>(plane_est, depth, ws);
    ph_finalize<<<1, 32, 0, stream>>>(ws, kinv, out);
}